// ModelSGLang_60533269069831
// MI455X (gfx1250) — compile-verified
//
#include <hip/hip_runtime.h>

// Token-table column count, fixed by the reference harness (CTX = 65536).
#define CTX_COLS 65536

// ---------------------------------------------------------------------------
// Kernel 1: build row->batch inverse map and exclusive cumsum of extend_lens.
// Single block; B and ROWS are tiny (128 / 512).
// ---------------------------------------------------------------------------
__global__ void setup_maps(const int* __restrict__ pool_idx,
                           const int* __restrict__ extend_lens,
                           int* __restrict__ inv,        // [rows]
                           int* __restrict__ ext_start,  // [b_count]
                           int rows, int b_count) {
  const int t = threadIdx.x;
  for (int r = t; r < rows; r += blockDim.x) inv[r] = -1;
  __syncthreads();
  for (int b = t; b < b_count; b += blockDim.x) {
    int s = 0;
    for (int i = 0; i < b; ++i) s += extend_lens[i];  // B=128: negligible
    ext_start[b] = s;
    inv[pool_idx[b]] = b;
  }
}

// ---------------------------------------------------------------------------
// Kernel 2: produce the updated req_to_token table.
// One int4 (16B) per thread; wave32 moves 512 contiguous bytes per b128 op.
// grid = (CTX/1024, ROWS), block = 256.
// ---------------------------------------------------------------------------
__global__ void __launch_bounds__(256)
write_req_to_token(const int* __restrict__ r2t,
                   const int* __restrict__ inv,
                   const int* __restrict__ prefix,
                   const int* __restrict__ prefix_lens,
                   const int* __restrict__ seq_lens,
                   const int* __restrict__ ext_start,
                   const int* __restrict__ out_cache,
                   int* __restrict__ out,
                   int maxp, int n_ocl) {
  const int row    = blockIdx.y;
  const int c0     = (blockIdx.x * 256 + threadIdx.x) * 4;
  const size_t idx = (size_t)row * CTX_COLS + (size_t)c0;
  const int b      = inv[row];  // uniform per block (row is SGPR)

  int4 v;
  if (b < 0) {
    // Untouched row: straight b128 copy.
    v = *(const int4*)(r2t + idx);
  } else {
    const int pl = prefix_lens[b];
    const int sl = seq_lens[b];
    if (c0 + 3 < pl) {
      // Entire vec4 inside the prefix region: aligned b128 load.
      v = *(const int4*)(prefix + (size_t)b * (size_t)maxp + (size_t)c0);
    } else if (c0 >= sl) {
      // Entire vec4 past seq_len: keep old values, aligned b128 load.
      v = *(const int4*)(r2t + idx);
    } else {
      // Boundary / extend region (rare: <= ~4K of 64K cols per mapped row).
      // out_cache gather has arbitrary alignment -> scalar b32 loads.
      const int es = ext_start[b];
      int vv[4];
#pragma unroll
      for (int k = 0; k < 4; ++k) {
        const int c = c0 + k;
        int val;
        if (c < pl) {
          val = prefix[(size_t)b * (size_t)maxp + (size_t)c];
        } else if (c < sl) {
          int e = es + (c - pl);
          e = e < 0 ? 0 : (e >= n_ocl ? n_ocl - 1 : e);  // reference's clip
          val = out_cache[e];
        } else {
          val = r2t[idx + (size_t)k];
        }
        vv[k] = val;
      }
      v = make_int4(vv[0], vv[1], vv[2], vv[3]);
    }
  }
  // Full-cacheline coalesced b128 store.
  *(int4*)(out + idx) = v;
}

// ---------------------------------------------------------------------------
// Host-side launcher (graph-capture safe: only kernel launches on `stream`).
// Input order: req_to_token, req_pool_indices, prefix_tensors, prefix_lens,
//              seq_lens, extend_lens, out_cache_loc
// ---------------------------------------------------------------------------
extern "C" void kernel_launch(void* const* d_in, const int* in_sizes, int n_in,
                              void* d_out, int out_size, void* d_ws, size_t ws_size,
                              hipStream_t stream) {
  const int* r2t        = (const int*)d_in[0];
  const int* pool_idx   = (const int*)d_in[1];
  const int* prefix     = (const int*)d_in[2];
  const int* prefix_len = (const int*)d_in[3];
  const int* seq_len    = (const int*)d_in[4];
  const int* extend_len = (const int*)d_in[5];
  const int* out_cache  = (const int*)d_in[6];
  int* out = (int*)d_out;  // int32 output; harness types it as float* but bits match

  const int B    = in_sizes[1];
  const int MAXP = in_sizes[2] / B;
  const int ROWS = in_sizes[0] / CTX_COLS;
  const int NOCL = in_sizes[6];

  int* inv       = (int*)d_ws;   // [ROWS]
  int* ext_start = inv + ROWS;   // [B]

  setup_maps<<<1, 256, 0, stream>>>(pool_idx, extend_len, inv, ext_start, ROWS, B);

  dim3 grid(CTX_COLS / (256 * 4), ROWS);  // (64, 512)
  write_req_to_token<<<grid, 256, 0, stream>>>(r2t, inv, prefix, prefix_len,
                                               seq_len, ext_start, out_cache,
                                               out, MAXP, NOCL);
}